// TestSparseConv_7447473292097
// MI455X (gfx1250) — compile-verified
//
#include <hip/hip_runtime.h>

#define SGRID 128
#define NVOX (2 * SGRID * SGRID * SGRID)   // B=2 batches of 128^3
#define C 64                                // C_in == C_out == 64
#define KP 32                               // K-pairs per 64-deep slab
#define PITCH 160                           // dwords per K-pair row (==32 mod 64 -> bank-clean)
#define SLAB (KP * PITCH)                   // 5120 floats = 20KB per tap

typedef __attribute__((ext_vector_type(2))) float v2f;
typedef __attribute__((ext_vector_type(8))) float v8f;

// 32-bit LDS byte address of a generic pointer into __shared__ space
__device__ inline unsigned lds_addr_of(const void* p) {
  return (unsigned)(size_t)(__attribute__((address_space(3))) const void*)p;
}

// async DMA one 20KB transposed weight slab into LDS: 5 x b128 per thread
__device__ inline void stage_weights_async(const float* __restrict__ slab,
                                           unsigned ldsDst, int tid) {
  const float* g = slab + tid * 4;
  const unsigned l = ldsDst + (unsigned)tid * 16u;
#pragma unroll
  for (int j = 0; j < 5; ++j) {
    asm volatile("global_load_async_to_lds_b128 %0, %1, off"
                 :: "v"(l + (unsigned)(j * 4096)), "v"(g + j * 1024)
                 : "memory");
  }
}

// ---------------- one-time weight transpose ----------------
// Wt[o][kp*PITCH + 2c + (k&1)] = W[o][k][c]  (K-pair interleave, padded pitch)

__global__ __launch_bounds__(256) void k_wt(const float* __restrict__ W,
                                            float* __restrict__ Wt) {
  int i = blockIdx.x * blockDim.x + threadIdx.x;
  if (i >= 27 * C * C) return;
  int c = i & 63;
  int k = (i >> 6) & 63;
  int o = i >> 12;
  Wt[(size_t)o * SLAB + (k >> 1) * PITCH + c * 2 + (k & 1)] = W[i];
}

// ---------------- neighbor-map construction ----------------

__global__ __launch_bounds__(256) void k_init_table(int* __restrict__ table, int n) {
  int i = blockIdx.x * blockDim.x + threadIdx.x;
  if (i < n) table[i] = -1;
}

__global__ __launch_bounds__(64) void k_zero_row(float* __restrict__ zrow) {
  zrow[threadIdx.x] = 0.0f;
}

__global__ __launch_bounds__(256) void k_scatter(const int* __restrict__ coords,
                                                 int* __restrict__ table, int N) {
  int n = blockIdx.x * blockDim.x + threadIdx.x;
  if (n >= N) return;
  int b = coords[4 * n + 0], x = coords[4 * n + 1];
  int y = coords[4 * n + 2], z = coords[4 * n + 3];
  table[((b * SGRID + x) * SGRID + y) * SGRID + z] = n;
}

__global__ __launch_bounds__(256) void k_nbr(const int* __restrict__ coords,
                                             const int* __restrict__ table,
                                             int* __restrict__ nbr, int N) {
  int n = blockIdx.x * blockDim.x + threadIdx.x;
  if (n >= N) return;
  int b = coords[4 * n + 0], x = coords[4 * n + 1];
  int y = coords[4 * n + 2], z = coords[4 * n + 3];
#pragma unroll
  for (int dx = -1; dx <= 1; ++dx)
#pragma unroll
    for (int dy = -1; dy <= 1; ++dy)
#pragma unroll
      for (int dz = -1; dz <= 1; ++dz) {
        int o = (dx + 1) * 9 + (dy + 1) * 3 + (dz + 1);  // matches _OFFS order
        int qx = x + dx, qy = y + dy, qz = z + dz;
        int pt = -1;
        if ((unsigned)qx < (unsigned)SGRID && (unsigned)qy < (unsigned)SGRID &&
            (unsigned)qz < (unsigned)SGRID)
          pt = table[((b * SGRID + qx) * SGRID + qy) * SGRID + qz];
        nbr[o * N + n] = pt;
      }
}

// ---------------- sparse conv as 27 gathered GEMMs (f32 WMMA) ----------------
// Wave: 16 points x 64 out-channels = 4 accumulator tiles (16x16 f32).
// Block: 8 waves. Transposed weights double-buffered in LDS via async DMA;
// B fragments are single ds_load_b64 at base+imm offsets (bank-conflict-free).

__global__ __launch_bounds__(256) void k_spconv(const float* __restrict__ feats,
                                                const float* __restrict__ Wt,   // [27][SLAB]
                                                const int* __restrict__ nbr,    // [27,N]
                                                const float* __restrict__ zrow, // [64] zeros
                                                float* __restrict__ out,        // [N,64]
                                                int N) {
  __shared__ float sW[2][SLAB];

  const int tid   = threadIdx.x;
  const int lane  = tid & 31;
  const int wave  = tid >> 5;
  const int row   = lane & 15;   // M index (A/C) and N index (B/C)
  const int khalf = lane >> 4;   // which K-pair of the 16x4 A fragment
  const int tileBase = (blockIdx.x * 8 + wave) * 16;
  const int gRow  = tileBase + row;
  const int nclmp = gRow < N ? gRow : N - 1;     // safe nbr row for tail tiles
  const bool rowOK = gRow < N;

  const unsigned ldsW0 = lds_addr_of(sW[0]);
  const unsigned ldsW1 = lds_addr_of(sW[1]);

  v8f acc[4] = {v8f{}, v8f{}, v8f{}, v8f{}};

  // prologue: stage tap 0 into buffer 0
  stage_weights_async(Wt, ldsW0, tid);
  asm volatile("s_wait_asynccnt 0" ::: "memory");
  __syncthreads();

  for (int o = 0; o < 27; ++o) {
    const int cur = o & 1;

    // prefetch next tap's slab into the other buffer (hidden behind WMMAs)
    if (o + 1 < 27)
      stage_weights_async(Wt + (size_t)(o + 1) * SLAB, cur ? ldsW0 : ldsW1, tid);

    // gather source row for this tap; absent neighbor -> zero row pointer
    int pt = nbr[o * N + nclmp];
    if (!rowOK) pt = -1;
    const float* src   = (pt >= 0) ? (feats + (size_t)pt * C) : zrow;
    const float* bbase = sW[cur] + khalf * PITCH + row * 2;  // + kb*2*PITCH + nt*32

#pragma unroll
    for (int kb = 0; kb < 16; ++kb) {       // K = 64 in steps of 4
      v2f a = *(const v2f*)(src + kb * 4 + khalf * 2);
#pragma unroll
      for (int nt = 0; nt < 4; ++nt) {      // 4 N-tiles of 16 out-channels
        v2f bf = *(const v2f*)(bbase + kb * 2 * PITCH + nt * 32);
        acc[nt] = __builtin_amdgcn_wmma_f32_16x16x4_f32(
            /*neg_a=*/false, a, /*neg_b=*/false, bf,
            /*c_mod=*/(short)0, acc[nt], /*reuse_a=*/false, /*reuse_b=*/false);
      }
    }

    // next tap's DMA complete (own wave) + all waves done reading this buffer
    asm volatile("s_wait_asynccnt 0" ::: "memory");
    __syncthreads();
  }

  // C/D layout: VGPR i -> M = i + 8*(lane>=16), N = lane%16
#pragma unroll
  for (int nt = 0; nt < 4; ++nt) {
#pragma unroll
    for (int i = 0; i < 8; ++i) {
      int gr = tileBase + i + 8 * khalf;
      if (gr < N) out[(size_t)gr * C + nt * 16 + row] = acc[nt][i];
    }
  }
}

// ---------------- launch ----------------

extern "C" void kernel_launch(void* const* d_in, const int* in_sizes, int n_in,
                              void* d_out, int out_size, void* d_ws, size_t ws_size,
                              hipStream_t stream) {
  const float* feats  = (const float*)d_in[0];
  const int*   coords = (const int*)d_in[1];
  const float* W1     = (const float*)d_in[2];
  const float* W2     = (const float*)d_in[3];
  float*       out    = (float*)d_out;

  const int N = in_sizes[1] / 4;  // coords is [N,4]

  // workspace layout (256B aligned slices)
  char*  ws    = (char*)d_ws;
  size_t off   = 0;
  int*   table = (int*)(ws + off);   off += (size_t)NVOX * 4;            // 16.8 MB
  off = (off + 255) & ~(size_t)255;
  int*   nbr   = (int*)(ws + off);   off += (size_t)27 * N * 4;          // 10.8 MB
  off = (off + 255) & ~(size_t)255;
  float* zrow  = (float*)(ws + off); off += 64 * 4;                      // 256 B zeros
  off = (off + 255) & ~(size_t)255;
  float* Wt1   = (float*)(ws + off); off += (size_t)27 * SLAB * 4;       // 540 KB
  off = (off + 255) & ~(size_t)255;
  float* Wt2   = (float*)(ws + off); off += (size_t)27 * SLAB * 4;       // 540 KB
  off = (off + 255) & ~(size_t)255;
  float* h     = (float*)(ws + off);                                     // 25.6 MB

  k_init_table<<<(NVOX + 255) / 256, 256, 0, stream>>>(table, NVOX);
  k_zero_row<<<1, 64, 0, stream>>>(zrow);
  k_scatter<<<(N + 255) / 256, 256, 0, stream>>>(coords, table, N);
  k_nbr<<<(N + 255) / 256, 256, 0, stream>>>(coords, table, nbr, N);

  const int wtElems = 27 * C * C;
  k_wt<<<(wtElems + 255) / 256, 256, 0, stream>>>(W1, Wt1);
  k_wt<<<(wtElems + 255) / 256, 256, 0, stream>>>(W2, Wt2);

  const int tiles  = (N + 15) / 16;
  const int blocks = (tiles + 7) / 8;
  k_spconv<<<blocks, 256, 0, stream>>>(feats, Wt1, nbr, zrow, h,   N);
  k_spconv<<<blocks, 256, 0, stream>>>(h,     Wt2, nbr, zrow, out, N);
}